// GNNFuse_31121333027282
// MI455X (gfx1250) — compile-verified
//
#include <hip/hip_runtime.h>
#include <math.h>

#define Cc 256
#define Hh 4
#define NEG_SLOPE 0.2f
#define EPS 1e-5f
#define HW 4096     // 64*64
#define Bb 32
#define Ee 13

typedef __attribute__((ext_vector_type(2))) float v2f;
typedef __attribute__((ext_vector_type(8))) float v8f;

__constant__ int c_SRC[Ee] = {1, 1, 2, 2, 3, 3, 1, 2, 3, 0, 1, 2, 3};
__constant__ int c_DST[Ee] = {2, 3, 1, 3, 1, 2, 0, 0, 0, 0, 1, 2, 3};

// ---- block-wide sum over 256 threads (8 wave32), broadcast to all threads ----
__device__ __forceinline__ float blockSum256(float v, float* red8, float* bc) {
    #pragma unroll
    for (int o = 16; o > 0; o >>= 1) v += __shfl_down(v, o, 32);
    const int lane = threadIdx.x & 31, w = threadIdx.x >> 5;
    __syncthreads();                 // protect red8/bc reuse
    if (lane == 0) red8[w] = v;
    __syncthreads();
    if (threadIdx.x == 0) {
        float s = 0.f;
        #pragma unroll
        for (int i = 0; i < 8; ++i) s += red8[i];
        *bc = s;
    }
    __syncthreads();
    return *bc;
}

// ---------------------------------------------------------------------------
// 1) Global average pool of x_ful / rgb / dep -> node features [B,4,C].
//    Node 0 = tok broadcast, 1 = pool(x_ful), 2 = pool(rgb), 3 = pool(dep).
//    grid = B*C blocks, 256 threads; each thread streams 16 floats per tensor.
// ---------------------------------------------------------------------------
__global__ void pool_kernel(const float* __restrict__ xf,
                            const float* __restrict__ rgb,
                            const float* __restrict__ dep,
                            const float* __restrict__ tok,
                            float* __restrict__ G /* [B*4, C] */) {
    __shared__ float red8[8];
    __shared__ float bc;
    const int b = blockIdx.x >> 8;
    const int c = blockIdx.x & 255;
    const int t = threadIdx.x;
    const size_t base = (size_t)(b * Cc + c) * HW;

    const float4* f4 = (const float4*)(xf + base);
    const float4* r4 = (const float4*)(rgb + base);
    const float4* d4 = (const float4*)(dep + base);
    float s0 = 0.f, s1 = 0.f, s2 = 0.f;
    #pragma unroll
    for (int i = 0; i < 4; ++i) {
        float4 a = f4[t + 256 * i]; s0 += a.x + a.y + a.z + a.w;
        float4 bv = r4[t + 256 * i]; s1 += bv.x + bv.y + bv.z + bv.w;
        float4 cv = d4[t + 256 * i]; s2 += cv.x + cv.y + cv.z + cv.w;
    }
    const float inv = 1.0f / (float)HW;
    float m0 = blockSum256(s0, red8, &bc) * inv;
    float m1 = blockSum256(s1, red8, &bc) * inv;
    float m2 = blockSum256(s2, red8, &bc) * inv;
    if (t == 0) {
        G[(b * 4 + 0) * Cc + c] = tok[c];
        G[(b * 4 + 1) * Cc + c] = m0;
        G[(b * 4 + 2) * Cc + c] = m1;
        G[(b * 4 + 3) * Cc + c] = m2;
    }
}

// ---------------------------------------------------------------------------
// 2) XP = G @ W : [128,256] x [256,1024] -> [128,1024] via V_WMMA_F32_16X16X4_F32
//    grid = 64 blocks (N tiles), 256 threads = 8 waves (M tiles 0..7).
//    A frag (16x4 f32): lane l, vgpr v -> M=l&15, K=kb+(l>>4)*2+v  (float2 load)
//    B frag (4x16 f32): lane l, vgpr v -> K=kb+(l>>4)*2+v, N=l&15
//    D (16x16 f32):     vgpr r, lanes0-15 -> M=r, lanes16-31 -> M=r+8
// ---------------------------------------------------------------------------
__global__ void gemm_xp(const float* __restrict__ X,  // [128,256]
                        const float* __restrict__ W,  // [256,1024]
                        float* __restrict__ XP) {     // [128,1024]
    const int lane = threadIdx.x & 31;
    const int wave = threadIdx.x >> 5;
    const int row0 = wave * 16;
    const int col0 = blockIdx.x * 16;
    const int m  = lane & 15;
    const int kq = (lane >> 4) * 2;

    v8f acc = {};
    const float* ap0 = X + (row0 + m) * Cc + kq;
    const float* bp0 = W + kq * (Hh * Cc) + col0 + m;
    for (int kb = 0; kb < Cc; kb += 4) {
        v2f a, b;
        const float* ap = ap0 + kb;
        a.x = ap[0];
        a.y = ap[1];
        const float* bp = bp0 + kb * (Hh * Cc);
        b.x = bp[0];
        b.y = bp[Hh * Cc];
        acc = __builtin_amdgcn_wmma_f32_16x16x4_f32(
            /*neg_a=*/false, a, /*neg_b=*/false, b,
            /*c_mod=*/(short)0, acc, /*reuse_a=*/false, /*reuse_b=*/false);
    }
    const int rbase = row0 + (lane >> 4) * 8;
    const int cidx  = col0 + m;
    #pragma unroll
    for (int r = 0; r < 8; ++r)
        XP[(rbase + r) * (Hh * Cc) + cidx] = acc[r];
}

// ---------------------------------------------------------------------------
// 3) Attention + aggregation + head-mean + bias + residual + LayerNorm + ReLU.
//    grid = B blocks, 256 threads (thread = channel).
// ---------------------------------------------------------------------------
__global__ void gat_attn(const float* __restrict__ XP,   // [128, 1024]
                         float* __restrict__ G,          // [128, 256] in/out
                         const float* __restrict__ asrc, // [4,256]
                         const float* __restrict__ adst, // [4,256]
                         const float* __restrict__ bias,
                         const float* __restrict__ gamma,
                         const float* __restrict__ beta) {
    __shared__ float xs[4][Hh][Cc];   // 16 KB
    __shared__ float es[4][Hh], ed[4][Hh];
    __shared__ float alpha[Ee][Hh];
    __shared__ float red8[8];
    __shared__ float bc;

    const int b = blockIdx.x;
    const int t = threadIdx.x;

    #pragma unroll
    for (int n = 0; n < 4; ++n)
        #pragma unroll
        for (int h = 0; h < Hh; ++h)
            xs[n][h][t] = XP[(b * 4 + n) * (Hh * Cc) + h * Cc + t];
    __syncthreads();

    // per-(node,head) attention logit halves
    if (t < 16) {
        const int n = t >> 2, h = t & 3;
        float s = 0.f, d = 0.f;
        for (int c = 0; c < Cc; ++c) {
            const float v = xs[n][h][c];
            s += v * asrc[h * Cc + c];
            d += v * adst[h * Cc + c];
        }
        es[n][h] = s;
        ed[n][h] = d;
    }
    __syncthreads();

    // segment softmax over incoming edges per (dst node, head)
    if (t < 16) {
        const int dn = t >> 2, h = t & 3;
        float ev[Ee];
        float mx = -INFINITY;
        for (int e = 0; e < Ee; ++e) {
            if (c_DST[e] == dn) {
                float x = es[c_SRC[e]][h] + ed[dn][h];
                x = (x > 0.f) ? x : NEG_SLOPE * x;
                ev[e] = x;
                mx = fmaxf(mx, x);
            }
        }
        float den = 0.f;
        for (int e = 0; e < Ee; ++e) {
            if (c_DST[e] == dn) {
                const float ex = __expf(ev[e] - mx);
                alpha[e][h] = ex;
                den += ex;
            }
        }
        const float rden = 1.0f / den;
        for (int e = 0; e < Ee; ++e)
            if (c_DST[e] == dn) alpha[e][h] *= rden;
    }
    __syncthreads();

    // aggregate messages, head mean, bias, residual, LayerNorm, ReLU
    for (int dn = 0; dn < 4; ++dn) {
        float acc = 0.f;
        #pragma unroll
        for (int e = 0; e < Ee; ++e) {
            if (c_DST[e] == dn) {
                const int sn = c_SRC[e];
                #pragma unroll
                for (int h = 0; h < Hh; ++h)
                    acc += alpha[e][h] * xs[sn][h][t];
            }
        }
        acc = acc * (1.0f / Hh) + bias[t] + G[(b * 4 + dn) * Cc + t];

        const float mu = blockSum256(acc, red8, &bc) * (1.0f / Cc);
        const float dv = acc - mu;
        const float var = blockSum256(dv * dv, red8, &bc) * (1.0f / Cc);
        float y = dv * rsqrtf(var + EPS) * gamma[t] + beta[t];
        y = fmaxf(y, 0.f);
        G[(b * 4 + dn) * Cc + t] = y;
        __syncthreads();
    }
}

// ---------------------------------------------------------------------------
// 4) out = x_ful * (1 + sigmoid(g[:,0,:]))  — coalesced float4 streaming.
// ---------------------------------------------------------------------------
__global__ void gate_kernel(const float* __restrict__ xf,
                            const float* __restrict__ G,
                            float* __restrict__ out) {
    const int b = blockIdx.x >> 8;
    const int c = blockIdx.x & 255;
    const int t = threadIdx.x;
    const float gv = G[(b * 4 + 0) * Cc + c];
    const float f = 1.0f + 1.0f / (1.0f + __expf(-gv));
    const size_t base = (size_t)(b * Cc + c) * HW;
    const float4* in4 = (const float4*)(xf + base);
    float4* o4 = (float4*)(out + base);
    #pragma unroll
    for (int i = 0; i < 4; ++i) {
        float4 v = in4[t + 256 * i];
        v.x *= f; v.y *= f; v.z *= f; v.w *= f;
        o4[t + 256 * i] = v;
    }
}

extern "C" void kernel_launch(void* const* d_in, const int* in_sizes, int n_in,
                              void* d_out, int out_size, void* d_ws, size_t ws_size,
                              hipStream_t stream) {
    const float* xf  = (const float*)d_in[0];
    const float* rgb = (const float*)d_in[1];
    const float* dep = (const float*)d_in[2];
    const float* tok = (const float*)d_in[3];

    float* G  = (float*)d_ws;      // [128,256] node features, 128 KiB
    float* XP = (float*)d_out;     // [128,1024] scratch inside d_out (512 KiB),
                                   // fully overwritten by gate_kernel at the end

    pool_kernel<<<Bb * Cc, 256, 0, stream>>>(xf, rgb, dep, tok, G);

    for (int layer = 0; layer < 2; ++layer) {
        const float* W  = (const float*)d_in[4 + 6 * layer + 0];
        const float* as = (const float*)d_in[4 + 6 * layer + 1];
        const float* ad = (const float*)d_in[4 + 6 * layer + 2];
        const float* bi = (const float*)d_in[4 + 6 * layer + 3];
        const float* ga = (const float*)d_in[4 + 6 * layer + 4];
        const float* be = (const float*)d_in[4 + 6 * layer + 5];

        gemm_xp<<<64, 256, 0, stream>>>(G, W, XP);
        gat_attn<<<Bb, 256, 0, stream>>>(XP, G, as, ad, bi, ga, be);
    }

    gate_kernel<<<Bb * Cc, 256, 0, stream>>>(xf, G, (float*)d_out);
}